// TraceableKernelizedAttention_4999341932652
// MI455X (gfx1250) — compile-verified
//
#include <hip/hip_runtime.h>

// ---------------------------------------------------------------------------
// Kernelized (linear) attention:  out = (phi(Q) @ KV) / (phi(Q) @ Z + eps)
//   Q : [4,32,4096,128] f32   KV : [128,128,128] f32   Z : [128,128,1] f32
// phi(x) = elu(x)+1 = (x>0 ? x+1 : exp(x))
//
// MI455X plan: memory-bound (~545MB @ 23.3TB/s ~= 23us); fp32 WMMA
// V_WMMA_F32_16X16X4_F32 keeps full reference precision with compute time
// near the bandwidth floor. KV tile lives in LDS in B-fragment order, Q
// strips staged through LDS in A-fragment order, all global IO coalesced.
// ---------------------------------------------------------------------------

typedef __attribute__((ext_vector_type(2))) float v2f;
typedef __attribute__((ext_vector_type(4))) float v4f;
typedef __attribute__((ext_vector_type(8))) float v8f;

#define BH 128
#define SEQ 4096
#define DIM 128

// LDS layout (in floats)
#define KVB_T_STRIDE 2056                 // per N-tile: 32 steps * 64 + 8 pad (8224 B)
#define KVB_FLOATS   (8 * KVB_T_STRIDE)   // 16448
#define QA_W_STRIDE  2112                 // per wave: 32 steps * 66 (264 B step -> no bank conflicts)
#define QA_FLOATS    (8 * QA_W_STRIDE)    // 16896
#define Z_FLOATS     128
#define SMEM_FLOATS  (KVB_FLOATS + QA_FLOATS + Z_FLOATS)   // 33472 floats = 133,888 B < 320 KB

__device__ __forceinline__ float lane_bcast(float v, int srcLane) {
    int iv = __builtin_bit_cast(int, v);
    int r  = __builtin_amdgcn_ds_bpermute(srcLane << 2, iv);   // wave32: bits [6:2]
    return __builtin_bit_cast(float, r);
}

__global__ __launch_bounds__(256) void lin_attn_wmma_f32(
    const float* __restrict__ Q, const float* __restrict__ KV,
    const float* __restrict__ Z, float* __restrict__ Out)
{
    __shared__ float smem[SMEM_FLOATS];
    float* kvb    = smem;                              // B fragments of KV
    float* qa_all = smem + KVB_FLOATS;                 // A fragments of phi(Q), per wave
    float* zs     = smem + KVB_FLOATS + QA_FLOATS;     // raw Z (128 floats)

    const int tid  = threadIdx.x;
    const int lane = tid & 31;
    const int wave = tid >> 5;
    const int h    = lane >> 4;        // half-wave: selects K sub-pair / M upper half
    const int m    = lane & 15;

    const int blk   = blockIdx.x;
    const int head  = blk >> 4;        // 16 row-chunks (of 256 rows) per head
    const int chunk = blk & 15;

    const float* Qh  = Q  + (size_t)head * SEQ * DIM + (size_t)chunk * 256 * DIM;
    const float* KVh = KV + (size_t)head * DIM * DIM;
    const float* Zh  = Z  + (size_t)head * DIM;
    float*       Oh  = Out + (size_t)head * SEQ * DIM + (size_t)chunk * 256 * DIM;

    if (tid < 128) zs[tid] = Zh[tid];

    // ---- stage KV into WMMA B-fragment order (once per block) --------------
    // element (K,N): step s=K/4, half hh=(K%4)/2, j=K%2, tile t=N/16, col mm=N%16
    // fragment lane = mm + 16*hh ; float offset = t*2056 + s*64 + lane*2 + j
    #pragma unroll
    for (int it = 0; it < 16; ++it) {
        int idx = (it * 256 + tid) * 4;                 // 4 consecutive elements, coalesced b128
        v4f v = *(const v4f*)(KVh + idx);
        int K  = idx >> 7;
        int n0 = idx & 127;
        int s  = K >> 2, hh = (K >> 1) & 1, j = K & 1;
        float* base = kvb + s * 64 + 32 * hh + j;
        #pragma unroll
        for (int e = 0; e < 4; ++e) {
            int n = n0 + e;
            base[(n >> 4) * KVB_T_STRIDE + (n & 15) * 2] = v[e];
        }
    }
    __syncthreads();

    float* qa = qa_all + wave * QA_W_STRIDE;
    const v2f* qaf = (const v2f*)qa;            // index: s*33 + lane
    const v2f* zf  = (const v2f*)zs;            // index: s*2 + h  (broadcast within half)
    const v2f* kf  = (const v2f*)kvb;           // index: t*1028 + s*32 + lane

    for (int sp = 0; sp < 2; ++sp) {
        const int row0 = (wave + sp * 8) * 16;  // 16-row strip inside the 256-row chunk

        // ---- stage phi(Q) strip into A-fragment order ----------------------
        // wave loads one row per b128 (fully coalesced 512B); thread lane owns
        // cols 4*lane..4*lane+3 == K-step s=lane, writes two b64 fragments.
        const float* Qs = Qh + (size_t)row0 * DIM;
        #pragma unroll
        for (int r = 0; r < 16; ++r) {
            v4f q = __builtin_nontemporal_load((const v4f*)(Qs + r * DIM + 4 * lane));
            v4f p;
            #pragma unroll
            for (int e = 0; e < 4; ++e)
                p[e] = q[e] > 0.0f ? q[e] + 1.0f : __expf(q[e]);
            v2f* dst = (v2f*)(qa + lane * 66);
            dst[r]      = (v2f){p.x, p.y};      // fragment lane r      (k = 4s, 4s+1)
            dst[r + 16] = (v2f){p.z, p.w};      // fragment lane r+16   (k = 4s+2, 4s+3)
        }

        // ---- 16x128 strip GEMM: 32 K-steps x 8 N-tiles of WMMA f32 ---------
        v8f acc[8];
        #pragma unroll
        for (int t = 0; t < 8; ++t) acc[t] = (v8f)0.0f;
        float dsum = 0.0f;

        #pragma unroll 4
        for (int s = 0; s < 32; ++s) {
            v2f a  = qaf[s * 33 + lane];
            v2f zv = zf[s * 2 + h];
            dsum += a.x * zv.x + a.y * zv.y;    // denominator folded in (VALU)
            #pragma unroll
            for (int t = 0; t < 8; ++t) {
                v2f b = kf[t * (KVB_T_STRIDE / 2) + s * 32 + lane];
                acc[t] = __builtin_amdgcn_wmma_f32_16x16x4_f32(
                    /*neg_a=*/false, a, /*neg_b=*/false, b,
                    /*c_mod=*/(short)0, acc[t],
                    /*reuse_a=*/false, /*reuse_b=*/false);
            }
        }

        // combine K halves of the denominator (lane <-> lane^16), invert once
        dsum += lane_bcast(dsum, lane ^ 16);
        float inv = 1.0f / (dsum + 1e-6f);

        // ---- scale by 1/denom and store ------------------------------------
        // C/D layout: vgpr r, lane -> element (M = r + 8h, N = t*16 + m)
        #pragma unroll
        for (int r = 0; r < 8; ++r) {
            float sc = lane_bcast(inv, r + 8 * h);   // inv[M] lives in lane M (0..15)
            float* orow = Oh + (size_t)(row0 + r + 8 * h) * DIM + m;
            #pragma unroll
            for (int t = 0; t < 8; ++t)
                __builtin_nontemporal_store(acc[t][r] * sc, orow + t * 16);
        }
    }
}

extern "C" void kernel_launch(void* const* d_in, const int* in_sizes, int n_in,
                              void* d_out, int out_size, void* d_ws, size_t ws_size,
                              hipStream_t stream) {
    const float* Q  = (const float*)d_in[0];
    const float* KV = (const float*)d_in[1];
    const float* Z  = (const float*)d_in[2];
    float* out = (float*)d_out;
    dim3 grid(BH * (SEQ / 256));   // 2048 workgroups
    lin_attn_wmma_f32<<<grid, dim3(256), 0, stream>>>(Q, KV, Z, out);
}